// GraphConv_27951647162602
// MI455X (gfx1250) — compile-verified
//
#include <hip/hip_runtime.h>

typedef float v2f __attribute__((ext_vector_type(2)));
typedef float v8f __attribute__((ext_vector_type(8)));

#define N_NODES 50000
#define N_EDGES 640000
#define FEAT 128
#define M_TILES ((N_NODES + 15) / 16)   // 3125

// --- Kernel 1: scatter-add neighbour features + counts ------------------
// 2 edges per 256-thread block; 128 lanes cover one feature row (512B,
// fully coalesced gather). Scatter via global_atomic_add_f32.
__global__ __launch_bounds__(256) void gc_scatter_kernel(
    const float* __restrict__ features,
    const int*   __restrict__ edges,     // [2, N_EDGES] int32
    float*       __restrict__ summed,    // [N_NODES, FEAT]
    float*       __restrict__ counts)    // [N_NODES]
{
    const int e = blockIdx.x * 2 + (threadIdx.x >> 7);
    const int t = threadIdx.x & 127;
    if (e >= N_EDGES) return;
    const int dst = edges[e];             // edges[0][e]
    const int src = edges[N_EDGES + e];   // edges[1][e]
    const float v = features[(size_t)src * FEAT + t];
    atomicAdd(&summed[(size_t)dst * FEAT + t], v);
    if (t == 0) atomicAdd(&counts[dst], 1.0f);
}

// --- Kernel 2: WMMA f32 GEMM (16x16x4), weight staged in LDS ------------
// LDS weight layout is K-pair interleaved + bank-swizzled:
//   W[k][n]  ->  ldsW[(k>>1)*256 + (n ^ (((k>>1)&1)<<4))*2 + (k&1)]
// so each lane's B operand {W[k][n], W[k+1][n]} is one aligned ds_load_b64,
// and the two half-waves (consecutive k-pairs) hit disjoint bank sets.
__global__ __launch_bounds__(256) void gc_gemm_kernel(
    const float* __restrict__ A,        // [N_NODES, FEAT] (features or summed)
    const float* __restrict__ W,        // [FEAT, FEAT] row-major (in, out)
    const float* __restrict__ counts,   // [N_NODES] (used if useScale)
    float*       __restrict__ out,      // [N_NODES, 2*FEAT]
    int colOffset, int useScale)
{
    __shared__ float ldsW[FEAT * FEAT];   // 64 KB

    // Cooperative swizzled stage of the weight matrix into LDS.
    // Each thread moves float4s: 4 consecutive n at fixed k.
    for (int i = threadIdx.x; i < FEAT * FEAT / 4; i += 256) {
        const float4 w4 = ((const float4*)W)[i];
        const int k     = i >> 5;            // i*4/128
        const int n0    = (i & 31) << 2;     // (i*4)%128
        const int kpair = k >> 1;
        const int sw    = (kpair & 1) << 4;
        float* dst = ldsW + kpair * 256 + (n0 ^ sw) * 2 + (k & 1);
        dst[0] = w4.x; dst[2] = w4.y; dst[4] = w4.z; dst[6] = w4.w;
    }
    __syncthreads();

    const int wave  = threadIdx.x >> 5;
    const int lane  = threadIdx.x & 31;
    const int mtile = blockIdx.x * 8 + wave;
    if (mtile >= M_TILES) return;         // tail waves idle (no more barriers)

    // ISA 7.12.2 layouts (wave32):
    //  A 16x4 f32 : lane&15 = row M, lane>>4 selects K-pair {0,1}/{2,3}
    //  B  4x16 f32: lane&15 = col N, lane>>4 selects K-pair
    //  C/D 16x16  : VGPR i -> M=i (lanes 0-15) / M=i+8 (lanes 16-31)
    const int row = mtile * 16 + (lane & 15);
    const int kb  = (lane >> 4) * 2;
    const int col = lane & 15;

    float scale = 1.0f;
    if (useScale) {
        const float c = counts[row];
        scale = (c > 0.0f) ? (1.0f / c) : 0.0f;   // empty segment -> 0
    }

    v8f acc[8] = {};
    const float* aptr = A + (size_t)row * FEAT;

#pragma unroll 4
    for (int k0 = 0; k0 < FEAT; k0 += 4) {
        const float2 av = *(const float2*)(aptr + k0 + kb);  // global_load_b64
        v2f a;
        a.x = av.x * scale;
        a.y = av.y * scale;
        const int kpair = (k0 + kb) >> 1;
        const int ksw   = kpair & 1;                 // column-block swizzle bit
        const float* brow = ldsW + kpair * 256 + col * 2;
#pragma unroll
        for (int j = 0; j < 8; ++j) {
            const int jj = j ^ ksw;                  // undo store-side swizzle
            const v2f b = *(const v2f*)(brow + jj * 32);   // ds_load_b64
            acc[j] = __builtin_amdgcn_wmma_f32_16x16x4_f32(
                false, a, false, b, (short)0, acc[j], false, false);
        }
    }

    // Fused ReLU store. Output row stride = 2*FEAT (concat layout).
    const int rowbase = mtile * 16 + (lane >> 4) * 8;
#pragma unroll
    for (int j = 0; j < 8; ++j) {
#pragma unroll
        for (int i = 0; i < 8; ++i) {
            const float v = acc[j][i];
            out[(size_t)(rowbase + i) * (2 * FEAT) + colOffset + j * 16 + col] =
                v > 0.0f ? v : 0.0f;
        }
    }
}

extern "C" void kernel_launch(void* const* d_in, const int* in_sizes, int n_in,
                              void* d_out, int out_size, void* d_ws, size_t ws_size,
                              hipStream_t stream) {
    const float* features = (const float*)d_in[0];
    const int*   edges    = (const int*)d_in[1];
    const float* weight   = (const float*)d_in[2];
    float*       out      = (float*)d_out;

    float* summed = (float*)d_ws;                       // [N_NODES, FEAT]
    float* counts = summed + (size_t)N_NODES * FEAT;    // [N_NODES]

    // Zero accumulation workspace every call (graph-capture safe).
    hipMemsetAsync(d_ws, 0,
                   ((size_t)N_NODES * FEAT + N_NODES) * sizeof(float), stream);

    // Edge scatter-mean accumulation.
    gc_scatter_kernel<<<(N_EDGES + 1) / 2, 256, 0, stream>>>(
        features, edges, summed, counts);

    // Half 0: relu(features @ W) -> cols [0,128)
    const dim3 gemmGrid((M_TILES + 7) / 8);
    gc_gemm_kernel<<<gemmGrid, 256, 0, stream>>>(
        features, weight, counts, out, /*colOffset=*/0, /*useScale=*/0);

    // Half 1: relu(mean_agg @ W) -> cols [128,256), mean fused into A-load
    gc_gemm_kernel<<<gemmGrid, 256, 0, stream>>>(
        summed, weight, counts, out, /*colOffset=*/128, /*useScale=*/1);
}